// Peaking_56538949485208
// MI455X (gfx1250) — compile-verified
//
#include <hip/hip_runtime.h>
#include <hip/hip_bf16.h>

// Peaking biquad IIR via block-parallel linear-recurrence decomposition.
// Phase 3 applies the rank-2 start-state correction with V_WMMA_F32_16X16X4_F32.

#define NBATCH 64
#define NSAMP  524288
#define CHUNK  256                 // samples per chunk (serial depth phase 1)
#define NCHUNK (NSAMP / CHUNK)     // 2048 chunks per batch (serial depth phase 2)

typedef __attribute__((ext_vector_type(2))) float v2f;
typedef __attribute__((ext_vector_type(8))) float v8f;

// ---------------- Phase 0: coefficients + homogeneous response p,q + A^L -----
__global__ void ph0_coeffs(const float* __restrict__ cp,
                           float* __restrict__ coeff,   // [NBATCH][16]
                           float* __restrict__ p,       // [NBATCH][CHUNK]
                           float* __restrict__ q) {     // [NBATCH][CHUNK]
  int b = threadIdx.x;
  if (b >= NBATCH) return;
  float gain_db = -12.0f + cp[b * 3 + 0] * 24.0f;
  float cutoff  =  20.0f + cp[b * 3 + 1] * (20000.0f - 20.0f);
  float Q       =   0.1f + cp[b * 3 + 2] * (10.0f - 0.1f);

  float A     = powf(10.0f, gain_db * (1.0f / 40.0f));
  float w0    = 6.2831853071795864f * cutoff / 44100.0f;
  float alpha = sinf(w0) / (2.0f * Q);
  float cw    = cosf(w0);
  float a0    = 1.0f + alpha / A;
  float inv   = 1.0f / a0;
  float b0 = (1.0f + alpha * A) * inv;
  float b1 = (-2.0f * cw) * inv;
  float b2 = (1.0f - alpha * A) * inv;
  float a1 = (-2.0f * cw) * inv;
  float a2 = (1.0f - alpha / A) * inv;

  float* cf = coeff + b * 16;
  cf[0] = b0; cf[1] = b1; cf[2] = b2; cf[3] = a1; cf[4] = a2;

  // p[k], q[k] = first row of A^k ; A = [[-a1, 1], [-a2, 0]]
  float m00 = 1.f, m01 = 0.f, m10 = 0.f, m11 = 1.f;
  for (int k = 0; k < CHUNK; ++k) {
    p[b * CHUNK + k] = m00;
    q[b * CHUNK + k] = m01;
    float n00 = -a1 * m00 + m10;
    float n01 = -a1 * m01 + m11;
    float n10 = -a2 * m00;
    float n11 = -a2 * m01;
    m00 = n00; m01 = n01; m10 = n10; m11 = n11;
  }
  cf[5] = m00; cf[6] = m01; cf[7] = m10; cf[8] = m11;   // A^CHUNK
}

// ---------------- Phase 1: per-chunk local recurrence (zero init state) ------
// Block = 256 threads = 8 waves; each wave owns 32 consecutive chunks.
// 32x32 LDS transpose tiles -> coalesced global traffic, conflict-free LDS.
__global__ __launch_bounds__(256) void ph1_local(
    const float* __restrict__ x, float* __restrict__ y,
    const float* __restrict__ coeff,
    float* __restrict__ e1, float* __restrict__ e2) {
  __shared__ float tile[8][32][33];
  int lane = threadIdx.x & 31;
  int wave = threadIdx.x >> 5;
  int chunk0 = blockIdx.x * 256 + wave * 32;   // first chunk handled by wave
  int b  = chunk0 / NCHUNK;                     // whole block in one batch
  int c0 = chunk0 % NCHUNK;
  const float* cf = coeff + b * 16;
  float b0 = cf[0], b1c = cf[1], b2c = cf[2], a1 = cf[3], a2 = cf[4];
  size_t base = (size_t)b * NSAMP;

  float z1 = 0.f, z2 = 0.f;                     // zero init state per chunk
  for (int j = 0; j < CHUNK; j += 32) {
    __syncthreads();
    #pragma unroll
    for (int r = 0; r < 32; ++r)                // coalesced loads
      tile[wave][r][lane] = x[base + (size_t)(c0 + r) * CHUNK + j + lane];
    __syncthreads();
    #pragma unroll
    for (int s = 0; s < 32; ++s) {              // lane's private recurrence
      float xv = tile[wave][lane][s];
      float yv = fmaf(b0, xv, z1);
      z1 = fmaf(b1c, xv, fmaf(-a1, yv, z2));
      z2 = fmaf(b2c, xv, -a2 * yv);
      tile[wave][lane][s] = yv;
    }
    __syncthreads();
    #pragma unroll
    for (int r = 0; r < 32; ++r)                // coalesced stores
      y[base + (size_t)(c0 + r) * CHUNK + j + lane] = tile[wave][r][lane];
  }
  e1[b * NCHUNK + c0 + lane] = z1;              // lane's chunk = c0+lane
  e2[b * NCHUNK + c0 + lane] = z2;
}

// ---------------- Phase 2: serial scan of chunk start states -----------------
__global__ void ph2_scan(const float* __restrict__ coeff,
                         const float* __restrict__ e1, const float* __restrict__ e2,
                         float* __restrict__ s1, float* __restrict__ s2) {
  int b = threadIdx.x;
  if (b >= NBATCH) return;
  const float* cf = coeff + b * 16;
  float AL00 = cf[5], AL01 = cf[6], AL10 = cf[7], AL11 = cf[8];
  float z1 = 0.f, z2 = 0.f;
  for (int c = 0; c < NCHUNK; ++c) {
    s1[b * NCHUNK + c] = z1;
    s2[b * NCHUNK + c] = z2;
    float n1 = fmaf(AL00, z1, fmaf(AL01, z2, e1[b * NCHUNK + c]));
    float n2 = fmaf(AL10, z1, fmaf(AL11, z2, e2[b * NCHUNK + c]));
    z1 = n1; z2 = n2;
  }
}

// ---------------- Phase 3: rank-2 correction via WMMA f32 16x16x4 ------------
// corr(16 chunks x 16 samples) = A(16x4: [z1s z2s 0 0]) @ B(4x16: rows p,q,0,0)
// D tile: M = chunk offset, N = sample offset.
// Addressing: one per-lane base pointer; the 8 C/D elements are at constant
// byte offsets v*CHUNK*4 (immediate IOFFSET), base advances 64B per k-tile.
// All guarded operands load from clamped addresses (lane&15) unconditionally
// and select zero afterwards -> no exec-mask branches.
__global__ __launch_bounds__(256) void ph3_wmma(
    float* __restrict__ y,
    const float* __restrict__ p, const float* __restrict__ q,
    const float* __restrict__ s1, const float* __restrict__ s2) {
  int lane = threadIdx.x & 31;
  int wave = threadIdx.x >> 5;
  int w  = blockIdx.x * 8 + wave;
  int b  = w / (NCHUNK / 16);
  int ct = w % (NCHUNK / 16);
  int c0 = ct * 16;
  int l15  = lane & 15;
  bool lo  = lane < 16;

  // A operand (16x4 f32 layout): lanes 0-15 = K0/K1 cols; lanes 16-31 = K2/K3 = 0
  float az1 = s1[b * NCHUNK + c0 + l15];
  float az2 = s2[b * NCHUNK + c0 + l15];
  v2f a;
  a.x = lo ? az1 : 0.f;
  a.y = lo ? az2 : 0.f;

  // per-lane pointers; k-loop indexes with constant offsets only
  const float* pp = p + b * CHUNK + l15;
  const float* qp = q + b * CHUNK + l15;
  int m = (lane >> 4) << 3;   // 0 or 8 (C/D row half per lane group)
  float* yp = y + (size_t)b * NSAMP + (size_t)(c0 + m) * CHUNK + l15;

  for (int k0 = 0; k0 < CHUNK; k0 += 16) {
    // B operand (4x16 f32 layout): lanes 0-15 = rows K0/K1; lanes 16-31 = rows 2,3 = 0
    float pv = pp[k0];
    float qv = qp[k0];
    v2f bb;
    bb.x = lo ? pv : 0.f;
    bb.y = lo ? qv : 0.f;

    v8f c;
    #pragma unroll
    for (int v = 0; v < 8; ++v)
      c[v] = yp[v * CHUNK + k0];        // immediate offsets v*1024 bytes

    c = __builtin_amdgcn_wmma_f32_16x16x4_f32(
        /*neg_a=*/false, a, /*neg_b=*/false, bb,
        /*c_mod=*/(short)0, c, /*reuse_a=*/false, /*reuse_b=*/false);

    #pragma unroll
    for (int v = 0; v < 8; ++v)
      yp[v * CHUNK + k0] = c[v];
  }
}

extern "C" void kernel_launch(void* const* d_in, const int* in_sizes, int n_in,
                              void* d_out, int out_size, void* d_ws, size_t ws_size,
                              hipStream_t stream) {
  const float* x  = (const float*)d_in[0];   // (64,1,524288)
  const float* cp = (const float*)d_in[1];   // (64,3,1)
  float* out = (float*)d_out;
  float* ws  = (float*)d_ws;

  float* coeff = ws;                          // 64*16
  float* p  = coeff + NBATCH * 16;            // 64*CHUNK
  float* q  = p + NBATCH * CHUNK;             // 64*CHUNK
  float* e1 = q + NBATCH * CHUNK;             // 64*NCHUNK
  float* e2 = e1 + NBATCH * NCHUNK;
  float* s1 = e2 + NBATCH * NCHUNK;
  float* s2 = s1 + NBATCH * NCHUNK;           // total ~2.2 MB of ws

  ph0_coeffs<<<1, 64, 0, stream>>>(cp, coeff, p, q);
  ph1_local<<<(NBATCH * NCHUNK) / 256, 256, 0, stream>>>(x, out, coeff, e1, e2);
  ph2_scan<<<1, 64, 0, stream>>>(coeff, e1, e2, s1, s2);
  ph3_wmma<<<(NBATCH * (NCHUNK / 16)) / 8, 256, 0, stream>>>(out, p, q, s1, s2);
}